// LinkAttention_36919538876765
// MI455X (gfx1250) — compile-verified
//
#include <hip/hip_runtime.h>
#include <hip/hip_bf16.h>
#include <math.h>

typedef __attribute__((ext_vector_type(2))) float v2f;
typedef __attribute__((ext_vector_type(8))) float v8f;

#define LROWS 262144
#define DDIM  128
#define HEADS 8
#define BGRAPH 1024
#define LDS_STRIDE 132   // 132 % 64 == 4 -> each of 16 rows hits a distinct 4-bank group

// ---------------- init: segmax=-inf, segsum=0, value=0 ----------------
__global__ __launch_bounds__(256) void init_kernel(float* __restrict__ segmax,
                                                   float* __restrict__ segsum,
                                                   float* __restrict__ value) {
    int i = blockIdx.x * 256 + threadIdx.x;
    if (i < BGRAPH * HEADS) { segmax[i] = -INFINITY; segsum[i] = 0.0f; }
    if (i < BGRAPH * DDIM)  { value[i] = 0.0f; }
}

// ---------------- pass 1: logits = x@Wq + bq via V_WMMA_F32_16X16X4_F32,
//                  store logits into score slot, atomic-fmax into segmax ----
__global__ __launch_bounds__(128) void logits_wmma_kernel(
    const float* __restrict__ x, const int* __restrict__ batch,
    const float* __restrict__ Wq, const float* __restrict__ bq,
    float* __restrict__ logitsOut, float* __restrict__ segmax)
{
    __shared__ float ldsWqT[16 * LDS_STRIDE];      // Wq^T padded to 16 heads
    __shared__ float ldsX[4][16 * LDS_STRIDE];     // 4 waves x (16 rows x 128 cols)

    const int tid  = threadIdx.x;
    const int w    = tid >> 5;
    const int lane = tid & 31;

    // Load Wq [128,8] transposed+padded into LDS: ldsWqT[n][k]
    for (int i = tid; i < 16 * 128; i += 128) {
        int n = i >> 7, k = i & 127;
        ldsWqT[n * LDS_STRIDE + k] = (n < HEADS) ? Wq[k * HEADS + n] : 0.0f;
    }

    const int rowBase = blockIdx.x * 64 + w * 16;
    // Coalesced float4 loads: one 512B row per iteration per wave
    const float4* xrow = (const float4*)(x + (size_t)rowBase * DDIM);
#pragma unroll
    for (int r = 0; r < 16; ++r) {
        float4 v = xrow[r * 32 + lane];
        *(float4*)&ldsX[w][r * LDS_STRIDE + lane * 4] = v;
    }
    __syncthreads();

    const int nm = lane & 15;                 // A: row m / B: col n / C: col n
    const int kb = (lane < 16) ? 0 : 2;       // K sub-offset for A/B fragments
    const float* Arow = &ldsX[w][nm * LDS_STRIDE];
    const float* Brow = &ldsWqT[nm * LDS_STRIDE];

    v8f c = {};
#pragma unroll
    for (int k = 0; k < 128; k += 4) {
        v2f a, b;
        a.x = Arow[k + kb]; a.y = Arow[k + kb + 1];
        b.x = Brow[k + kb]; b.y = Brow[k + kb + 1];
        c = __builtin_amdgcn_wmma_f32_16x16x4_f32(false, a, false, b,
                                                  (short)0, c, false, false);
    }

    const bool headOK = (nm < HEADS);
    const float bias = headOK ? bq[nm] : 0.0f;
    const int moff = (lane < 16) ? 0 : 8;     // C rows: lanes 0-15 -> M=v, 16-31 -> M=v+8

    float mx = -INFINITY;
#pragma unroll
    for (int v = 0; v < 8; ++v) {
        float lg = c[v] + bias;
        c[v] = lg;
        mx = fmaxf(mx, lg);
        if (headOK)
            logitsOut[(size_t)(rowBase + moff + v) * HEADS + nm] = lg;
    }

    const int b0  = batch[rowBase];
    const int b15 = batch[rowBase + 15];
    if (b0 == b15) {
        // whole tile in one segment (common case): 8 atomics per tile
        float other = __shfl_xor(mx, 16, 32);
        mx = fmaxf(mx, other);
        if (lane < HEADS)
            __hip_atomic_fetch_max(&segmax[b0 * HEADS + lane], mx,
                                   __ATOMIC_RELAXED, __HIP_MEMORY_SCOPE_AGENT);
    } else if (headOK) {
#pragma unroll
        for (int v = 0; v < 8; ++v) {
            int bb = batch[rowBase + moff + v];
            __hip_atomic_fetch_max(&segmax[bb * HEADS + nm], c[v],
                                   __ATOMIC_RELAXED, __HIP_MEMORY_SCOPE_AGENT);
        }
    }
}

// ---------------- pass 2: e = exp(logit - segmax), block-reduced segsum ----
__global__ __launch_bounds__(256) void esum_kernel(
    const float* __restrict__ segmax, const int* __restrict__ batch,
    float* __restrict__ eInOut, float* __restrict__ segsum)
{
    __shared__ float red[256];
    const int tid = threadIdx.x;
    const size_t idx = (size_t)blockIdx.x * 256 + tid;
    const int r = (int)(idx >> 3);
    const int h = (int)(idx & 7);
    const int b = batch[r];

    float e = __expf(eInOut[idx] - segmax[b * HEADS + h]);
    eInOut[idx] = e;

    const int r0 = blockIdx.x * 32;
    if (batch[r0] == batch[r0 + 31]) {
        red[tid] = e;
        __syncthreads();
        for (int s = 128; s >= 8; s >>= 1) {
            if (tid < s) red[tid] += red[tid + s];
            __syncthreads();
        }
        if (tid < HEADS)
            __hip_atomic_fetch_add(&segsum[b * HEADS + tid], red[tid],
                                   __ATOMIC_RELAXED, __HIP_MEMORY_SCOPE_AGENT);
    } else {
        __hip_atomic_fetch_add(&segsum[b * HEADS + h], e,
                               __ATOMIC_RELAXED, __HIP_MEMORY_SCOPE_AGENT);
    }
}

// ---------------- pass 3: score = e/(sum+eps), per-row head-sum s[r] -------
__global__ __launch_bounds__(256) void score_kernel(
    const float* __restrict__ segsum, const int* __restrict__ batch,
    float* __restrict__ scoreInOut, float* __restrict__ sOut)
{
    const int r = blockIdx.x * 256 + threadIdx.x;
    const int b = batch[r];
    float s = 0.0f;
    float sc[HEADS];
#pragma unroll
    for (int h = 0; h < HEADS; ++h) {
        float e = scoreInOut[(size_t)r * HEADS + h];
        float v = e / (segsum[b * HEADS + h] + 1e-16f);
        sc[h] = v;
        s += v;
    }
#pragma unroll
    for (int h = 0; h < HEADS; ++h)
        scoreInOut[(size_t)r * HEADS + h] = sc[h];
    sOut[r] = s;
}

// ---------------- pass 4: value[b,d] += x[r,d]*s[r], sorted-segment flush --
__global__ __launch_bounds__(128) void value_kernel(
    const float* __restrict__ x, const int* __restrict__ batch,
    const float* __restrict__ s, float* __restrict__ value)
{
    const int d  = threadIdx.x;
    const int r0 = blockIdx.x * 256;
    float acc = 0.0f;
    int curb = batch[r0];
    for (int i = 0; i < 256; ++i) {
        const int r = r0 + i;
        const int b = batch[r];          // block-uniform -> scalar load
        if (b != curb) {
            __hip_atomic_fetch_add(&value[(size_t)curb * DDIM + d], acc,
                                   __ATOMIC_RELAXED, __HIP_MEMORY_SCOPE_AGENT);
            acc = 0.0f;
            curb = b;
        }
        acc = fmaf(x[(size_t)r * DDIM + d], s[r], acc);
    }
    __hip_atomic_fetch_add(&value[(size_t)curb * DDIM + d], acc,
                           __ATOMIC_RELAXED, __HIP_MEMORY_SCOPE_AGENT);
}

extern "C" void kernel_launch(void* const* d_in, const int* in_sizes, int n_in,
                              void* d_out, int out_size, void* d_ws, size_t ws_size,
                              hipStream_t stream) {
    const float* x     = (const float*)d_in[0];
    const int*   batch = (const int*)d_in[1];
    const float* Wq    = (const float*)d_in[2];
    const float* bq    = (const float*)d_in[3];

    float* out   = (float*)d_out;
    float* value = out;                       // [B, D]
    float* score = out + BGRAPH * DDIM;       // [L, H] (also holds logits, then e)

    float* segmax = (float*)d_ws;             // [B, H]
    float* segsum = segmax + BGRAPH * HEADS;  // [B, H]
    float* srow   = segsum + BGRAPH * HEADS;  // [L]

    init_kernel<<<(BGRAPH * DDIM) / 256, 256, 0, stream>>>(segmax, segsum, value);
    logits_wmma_kernel<<<LROWS / 64, 128, 0, stream>>>(x, batch, Wq, bq, score, segmax);
    esum_kernel<<<LROWS / 32, 256, 0, stream>>>(segmax, batch, score, segsum);
    score_kernel<<<LROWS / 256, 256, 0, stream>>>(segsum, batch, score, srow);
    value_kernel<<<LROWS / 256, 128, 0, stream>>>(x, batch, srow, value);
}